// RecurrentTransformation_59433757442155
// MI455X (gfx1250) — compile-verified
//
#include <hip/hip_runtime.h>
#include <stdint.h>

// Problem constants (match reference)
#define DIM   4096
#define HID   8192
#define ALPHA 0.1f

// Serial-phase config
#define PIPE      4            // async-to-LDS pipeline depth (4 x 32KB = 128KB of 320KB LDS)
#define R_THREADS 1024         // 32 waves (wave32)
#define R_WAVES   32

typedef __attribute__((ext_vector_type(2))) float v2f;
typedef __attribute__((ext_vector_type(8))) float v8f;
typedef __attribute__((ext_vector_type(4))) int   v4i;

#define AS_GLOBAL __attribute__((address_space(1)))
#define AS_LDS    __attribute__((address_space(3)))

// ---------------------------------------------------------------------------
// CDNA5 async copy: global -> LDS, 16B per lane, tracked with ASYNCcnt.
// ROCm builtin signature: (int4 AS1* gaddr, int4 AS3* lds, imm offset, imm cpol)
// ---------------------------------------------------------------------------
__device__ __forceinline__ void async_b128(float* lds_dst, const float* gsrc) {
#if __has_builtin(__builtin_amdgcn_global_load_async_to_lds_b128)
  AS_GLOBAL v4i* g = (AS_GLOBAL v4i*)(AS_GLOBAL float*)(float*)gsrc;
  AS_LDS    v4i* l = (AS_LDS    v4i*)(AS_LDS    float*)lds_dst;
  __builtin_amdgcn_global_load_async_to_lds_b128(g, l, 0, 0);
#else
  unsigned loff = (unsigned)(uintptr_t)(AS_LDS float*)lds_dst;
  unsigned long long ga = (unsigned long long)(uintptr_t)gsrc;
  asm volatile("global_load_async_to_lds_b128 %0, %1, off"
               :: "v"(loff), "v"(ga) : "memory");
#endif
}

__device__ __forceinline__ void wait_async_pipe() {
  // keep (PIPE-1) rows * 2 ops/row in flight; row t is guaranteed complete
#if __has_builtin(__builtin_amdgcn_s_wait_asynccnt)
  __builtin_amdgcn_s_wait_asynccnt(2 * (PIPE - 1));
#else
  asm volatile("s_wait_asynccnt 6" ::: "memory");
#endif
}

__device__ __forceinline__ void wait_ds_done() {
  // make sure DS reads of the slot completed before async engine rewrites it
#if __has_builtin(__builtin_amdgcn_s_wait_dscnt)
  __builtin_amdgcn_s_wait_dscnt(0);
#else
  asm volatile("s_wait_dscnt 0" ::: "memory");
#endif
}

// ---------------------------------------------------------------------------
// Full 32-lane sum via one V_WMMA_F32_16X16X4_F32.
// A(16x4) layout: lanes0-15 {v0=K0,v1=K1}, lanes16-31 {v0=K2,v1=K3}.
// With A = {x,0} and B = all-ones (layout invariant), C[m][n] = x_m + x_{m+16}.
// C(16x16): vgpr i holds rows m=i (lanes0-15) and m=i+8 (lanes16-31); summing
// the 8 vgprs + one xor-16 shuffle yields the total in every lane.
// ---------------------------------------------------------------------------
__device__ __forceinline__ float wave_sum32(float x) {
#if __has_builtin(__builtin_amdgcn_wmma_f32_16x16x4_f32)
  v2f a; a[0] = x;    a[1] = 0.0f;
  v2f b; b[0] = 1.0f; b[1] = 1.0f;
  v8f acc = {};
  acc = __builtin_amdgcn_wmma_f32_16x16x4_f32(false, a, false, b,
                                              (short)0, acc, false, false);
  float s = ((acc[0] + acc[1]) + (acc[2] + acc[3])) +
            ((acc[4] + acc[5]) + (acc[6] + acc[7]));
  s += __shfl_xor(s, 16, 32);
  return s;
#else
  for (int m = 1; m < 32; m <<= 1) x += __shfl_xor(x, m, 32);
  return x;
#endif
}

// ---------------------------------------------------------------------------
// Kernel 1: transpose x[HID][DIM] -> xT[DIM][HID] (coalesced both sides).
// xT (128MB) lands in the 192MB L2, feeding the serial kernel from L2.
// ---------------------------------------------------------------------------
__global__ __launch_bounds__(256) void transpose_kernel(
    const float* __restrict__ in, float* __restrict__ out) {
  __shared__ float tile[32][33];
  const int tx = threadIdx.x, ty = threadIdx.y;
  const int t0 = blockIdx.x * 32;   // DIM tile
  const int j0 = blockIdx.y * 32;   // HID tile
#pragma unroll
  for (int i = 0; i < 32; i += 8)
    tile[ty + i][tx] = in[(size_t)(j0 + ty + i) * DIM + (t0 + tx)];
  __syncthreads();
#pragma unroll
  for (int i = 0; i < 32; i += 8)
    out[(size_t)(t0 + ty + i) * HID + (j0 + tx)] = tile[tx][ty + i];
}

// ---------------------------------------------------------------------------
// Kernel 2: the serialized scalar recurrence.
//   c_{t+1} = a + sum_j v_j * relu(u*xT[t][j] + c_t)   (c_0 = a, s_0 = 0)
//   A[t]    = b + dw_t,  dw_t = w . s_t  (A[0] = b)
// One workgroup, 1024 threads; each lane owns 8 of the 8192 elements.
// x rows are streamed via async-to-LDS, depth-PIPE pipeline.
// ---------------------------------------------------------------------------
__global__ __launch_bounds__(R_THREADS, 1) void recur_kernel(
    const float* __restrict__ xT, const float* __restrict__ w,
    const float* __restrict__ v,  const float* __restrict__ bptr,
    const float* __restrict__ uptr, const float* __restrict__ aptr,
    float* __restrict__ A) {
  __shared__ __align__(16) float xbuf[PIPE][HID];
  __shared__ float part[2][R_WAVES * 2];   // ping-pong {pw,pv} per wave

  const int tid  = threadIdx.x;
  const int lane = tid & 31;
  const int wave = tid >> 5;
  const int o1 = 4 * tid;            // floats 0..4095
  const int o2 = HID / 2 + 4 * tid;  // floats 4096..8191

  const float u0 = uptr[0], a0 = aptr[0], b0 = bptr[0];

  // hold each lane's 8 weights in registers for the whole recurrence
  const float4 w1 = *(const float4*)(w + o1);
  const float4 w2 = *(const float4*)(w + o2);
  const float4 v1 = *(const float4*)(v + o1);
  const float4 v2 = *(const float4*)(v + o2);

  // prologue: fill the pipeline
#pragma unroll
  for (int t = 0; t < PIPE; ++t) {
    const float* src = xT + (size_t)t * HID;
    async_b128(&xbuf[t][o1], src + o1);
    async_b128(&xbuf[t][o2], src + o2);
  }
  if (tid == 0) A[0] = b0;           // dw_0 = 0

  float c = a0;                      // dv_0 + a
  for (int t = 0; t < DIM; ++t) {
    const int s = t & (PIPE - 1);
    wait_async_pipe();               // row t resident in LDS

    const float4 xa = *(const float4*)&xbuf[s][o1];
    const float4 xb = *(const float4*)&xbuf[s][o2];

    float pw = 0.0f, pv = 0.0f, r;
    r = fmaxf(fmaf(u0, xa.x, c), 0.0f); pw = fmaf(w1.x, r, pw); pv = fmaf(v1.x, r, pv);
    r = fmaxf(fmaf(u0, xa.y, c), 0.0f); pw = fmaf(w1.y, r, pw); pv = fmaf(v1.y, r, pv);
    r = fmaxf(fmaf(u0, xa.z, c), 0.0f); pw = fmaf(w1.z, r, pw); pv = fmaf(v1.z, r, pv);
    r = fmaxf(fmaf(u0, xa.w, c), 0.0f); pw = fmaf(w1.w, r, pw); pv = fmaf(v1.w, r, pv);
    r = fmaxf(fmaf(u0, xb.x, c), 0.0f); pw = fmaf(w2.x, r, pw); pv = fmaf(v2.x, r, pv);
    r = fmaxf(fmaf(u0, xb.y, c), 0.0f); pw = fmaf(w2.y, r, pw); pv = fmaf(v2.y, r, pv);
    r = fmaxf(fmaf(u0, xb.z, c), 0.0f); pw = fmaf(w2.z, r, pw); pv = fmaf(v2.z, r, pv);
    r = fmaxf(fmaf(u0, xb.w, c), 0.0f); pw = fmaf(w2.w, r, pw); pv = fmaf(v2.w, r, pv);

    // slot s consumed (per-lane slice is self-owned) -> refill with row t+PIPE
    if (t + PIPE < DIM) {
      wait_ds_done();
      const float* src = xT + (size_t)(t + PIPE) * HID;
      async_b128(&xbuf[s][o1], src + o1);
      async_b128(&xbuf[s][o2], src + o2);
    }

    // stage 1: 32-lane reduce per wave via WMMA
    pw = wave_sum32(pw);
    pv = wave_sum32(pv);
    if (lane == 0) {
      part[t & 1][2 * wave]     = pw;
      part[t & 1][2 * wave + 1] = pv;
    }
    __syncthreads();

    // stage 2: every wave redundantly reduces the 32 wave-partials (no bcast)
    const float qw = part[t & 1][2 * lane];
    const float qv = part[t & 1][2 * lane + 1];
    const float dw = wave_sum32(qw);
    const float dv = wave_sum32(qv);

    c = dv + a0;
    if (tid == 0 && t + 1 < DIM) A[t + 1] = dw + b0;
  }
}

// ---------------------------------------------------------------------------
// Kernel 3: z[j,t] = leakyrelu(y*x[j,t] + A[t]); logdet[j] from neg-count.
// One block per row: fully coalesced float4 reads/writes, bandwidth bound.
// ---------------------------------------------------------------------------
__global__ __launch_bounds__(256) void zout_kernel(
    const float* __restrict__ x, const float* __restrict__ A,
    const float* __restrict__ yptr, float* __restrict__ z,
    float* __restrict__ ld) {
  const int j = blockIdx.x, tid = threadIdx.x;
  const float y0 = yptr[0];
  const float log_ay = logf(fabsf(y0));
  const float log_al = logf(ALPHA);

  const float* xr = x + (size_t)j * DIM;
  float*       zr = z + (size_t)j * DIM;

  int cnt = 0;
#pragma unroll
  for (int it = 0; it < 4; ++it) {
    const int col = it * 1024 + 4 * tid;
    const float4 xv = *(const float4*)(xr + col);
    const float4 av = *(const float4*)(A + col);
    float4 lin, zv;
    lin.x = fmaf(y0, xv.x, av.x); lin.y = fmaf(y0, xv.y, av.y);
    lin.z = fmaf(y0, xv.z, av.z); lin.w = fmaf(y0, xv.w, av.w);
    zv.x = lin.x >= 0.0f ? lin.x : ALPHA * lin.x; cnt += (lin.x < 0.0f);
    zv.y = lin.y >= 0.0f ? lin.y : ALPHA * lin.y; cnt += (lin.y < 0.0f);
    zv.z = lin.z >= 0.0f ? lin.z : ALPHA * lin.z; cnt += (lin.z < 0.0f);
    zv.w = lin.w >= 0.0f ? lin.w : ALPHA * lin.w; cnt += (lin.w < 0.0f);
    *(float4*)(zr + col) = zv;
  }

  for (int m = 16; m; m >>= 1) cnt += __shfl_xor(cnt, m, 32);
  __shared__ int wsum[8];
  if ((tid & 31) == 0) wsum[tid >> 5] = cnt;
  __syncthreads();
  if (tid == 0) {
    int tot = 0;
#pragma unroll
    for (int i = 0; i < 8; ++i) tot += wsum[i];
    ld[j] = (float)DIM * log_ay + (float)tot * log_al;
  }
}

// ---------------------------------------------------------------------------
extern "C" void kernel_launch(void* const* d_in, const int* in_sizes, int n_in,
                              void* d_out, int out_size, void* d_ws, size_t ws_size,
                              hipStream_t stream) {
  (void)in_sizes; (void)n_in; (void)out_size; (void)ws_size;
  const float* x = (const float*)d_in[0];
  const float* y = (const float*)d_in[1];
  const float* w = (const float*)d_in[2];
  const float* b = (const float*)d_in[3];
  const float* u = (const float*)d_in[4];
  const float* v = (const float*)d_in[5];
  const float* a = (const float*)d_in[6];

  float* xT = (float*)d_ws;                     // DIM*HID floats (128 MB)
  float* A  = xT + (size_t)DIM * HID;           // DIM floats
  float* z  = (float*)d_out;                    // HID*DIM floats
  float* ld = z + (size_t)HID * DIM;            // HID floats

  transpose_kernel<<<dim3(DIM / 32, HID / 32), dim3(32, 8), 0, stream>>>(x, xT);
  recur_kernel<<<1, R_THREADS, 0, stream>>>(xT, w, v, b, u, a, A);
  zout_kernel<<<HID, 256, 0, stream>>>(x, A, y, z, ld);
}